// SimpleGCNConv_1614907703908
// MI455X (gfx1250) — compile-verified
//
#include <hip/hip_runtime.h>

// Problem constants (fixed by the reference)
#define N_NODES 8192
#define N_EDGES 262144
#define D       128
#define MASK_WORDS_PER_ROW (N_NODES / 32)                 // 256 u32 per row
#define MASK_WORDS (N_NODES * MASK_WORDS_PER_ROW)         // 2,097,152 u32 = 8 MB

typedef __attribute__((ext_vector_type(2))) float v2f;
typedef __attribute__((ext_vector_type(8))) float v8f;

// ---------------------------------------------------------------------------
// Kernel 1: zero the adjacency bitmask
// ---------------------------------------------------------------------------
__global__ void gcn_zero_mask(unsigned int* __restrict__ mask) {
  for (int i = blockIdx.x * blockDim.x + threadIdx.x; i < MASK_WORDS;
       i += gridDim.x * blockDim.x) {
    mask[i] = 0u;
  }
}

// ---------------------------------------------------------------------------
// Kernel 2: set adjacency bits (atomicOr is idempotent -> dedupes duplicate
// edges exactly like jnp .at[].set(1.0), and is deterministic)
// ---------------------------------------------------------------------------
__global__ void gcn_set_edges(const long long* __restrict__ ei,
                              unsigned int* __restrict__ mask) {
  int e = blockIdx.x * blockDim.x + threadIdx.x;
  if (e >= N_EDGES) return;
  unsigned int s = (unsigned int)ei[e];            // edge_index[0][e] (src)
  unsigned int d = (unsigned int)ei[N_EDGES + e];  // edge_index[1][e] (dst)
  atomicOr(&mask[s * MASK_WORDS_PER_ROW + (d >> 5)], 1u << (d & 31u));
}

// ---------------------------------------------------------------------------
// Kernel 3: row-normalized aggregation.
// One block per node, 128 threads = one feature each. All threads walk the
// row's 256 mask words in ascending order (same addresses across the wave ->
// broadcast loads), gathering x[j][:] coalesced across the feature dim.
// Self-loop added on top; degree = popcount(row) + 1 (always > 0).
// Deterministic: fixed summation order.
// ---------------------------------------------------------------------------
__global__ void gcn_aggregate(const unsigned int* __restrict__ mask,
                              const float* __restrict__ x,
                              float* __restrict__ agg) {
  const int i = blockIdx.x;       // node
  const int t = threadIdx.x;      // feature 0..127
  const unsigned int* row = mask + (size_t)i * MASK_WORDS_PER_ROW;

  float acc = x[(size_t)i * D + t];   // self-loop (eye added on top)
  int deg = 1;
  for (int w = 0; w < MASK_WORDS_PER_ROW; ++w) {
    unsigned int bits = row[w];
    deg += __popc(bits);
    while (bits) {
      int b = __ffs(bits) - 1;
      bits &= bits - 1u;
      int j = (w << 5) + b;
      acc += x[(size_t)j * D + t];
    }
  }
  agg[(size_t)i * D + t] = acc / (float)deg;
}

// ---------------------------------------------------------------------------
// Kernel 4: Y = Agg @ W^T + b via V_WMMA_F32_16X16X4_F32 (fp32 matrix pipe,
// matches the fp32 reference precision class).
//
// Per ISA layouts (CDNA5 ISA):
//   A (16x4 f32):  lane L: row M = L&15; VGPR0 = K=(L<16?0:2), VGPR1 = K+1
//                  -> one float2 load from Agg[m][k + 2*(L>>4)]
//   B (4x16 f32):  lane L: col N = L&15; VGPR g holds K rows g and g+2
//                  -> B[k'][n] = W[n][k'] so one float2 load from
//                     W[n0+n][k + 2*(L>>4)] (W is [D_OUT][D_IN] row-major)
//   C/D (16x16):   VGPR r: lanes 0-15 -> M=r, lanes 16-31 -> M=r+8; N = L&15
// One wave per 16x16 output tile, 32 WMMA ops over K=128.
// ---------------------------------------------------------------------------
__global__ void gcn_linear_wmma(const float* __restrict__ A,   // [8192][128]
                                const float* __restrict__ W,   // [128][128]
                                const float* __restrict__ bias,// [128]
                                float* __restrict__ Y) {       // [8192][128]
  const int lane = threadIdx.x & 31;
  const int wave = threadIdx.x >> 5;
  const int tile = blockIdx.x * (blockDim.x >> 5) + wave;
  const int m0 = (tile >> 3) << 4;   // 512 M-tiles
  const int n0 = (tile & 7) << 4;    // 8 N-tiles

  const int l16  = lane & 15;
  const int koff = (lane >> 4) << 1;               // 0 or 2

  const float* aptr = A + (size_t)(m0 + l16) * D + koff;
  const float* bptr = W + (size_t)(n0 + l16) * D + koff;

  v8f c = {0.f, 0.f, 0.f, 0.f, 0.f, 0.f, 0.f, 0.f};
  #pragma unroll 8
  for (int k = 0; k < D; k += 4) {
    v2f a = *(const v2f*)(aptr + k);
    v2f b = *(const v2f*)(bptr + k);
    // 8 args: (neg_a, A, neg_b, B, c_mod, C, reuse_a, reuse_b)
    c = __builtin_amdgcn_wmma_f32_16x16x4_f32(
        /*neg_a=*/false, a, /*neg_b=*/false, b,
        /*c_mod=*/(short)0, c, /*reuse_a=*/false, /*reuse_b=*/false);
  }

  const float bv = bias[n0 + l16];
  const int mbase = m0 + ((lane >> 4) << 3);       // +8 rows for upper half
  #pragma unroll
  for (int r = 0; r < 8; ++r) {
    Y[(size_t)(mbase + r) * D + n0 + l16] = c[r] + bv;
  }
}

// ---------------------------------------------------------------------------
// Launch. ws layout: [0, 8MB) adjacency bitmask; [8MB, 12MB) agg buffer.
// ---------------------------------------------------------------------------
extern "C" void kernel_launch(void* const* d_in, const int* in_sizes, int n_in,
                              void* d_out, int out_size, void* d_ws, size_t ws_size,
                              hipStream_t stream) {
  const float* x       = (const float*)d_in[0];      // [8192][128] f32
  const long long* ei  = (const long long*)d_in[1];  // [2][262144] i64
  const float* W       = (const float*)d_in[2];      // [128][128] f32
  const float* b       = (const float*)d_in[3];      // [128] f32
  float* Y             = (float*)d_out;              // [8192][128] f32

  unsigned int* mask = (unsigned int*)d_ws;
  float* agg = (float*)((char*)d_ws + (size_t)MASK_WORDS * sizeof(unsigned int));

  gcn_zero_mask<<<1024, 256, 0, stream>>>(mask);
  gcn_set_edges<<<N_EDGES / 256, 256, 0, stream>>>(ei, mask);
  gcn_aggregate<<<N_NODES, 128, 0, stream>>>(mask, x, agg);

  // 4096 tiles, 4 waves (128 threads) per block -> 1024 blocks
  gcn_linear_wmma<<<(N_NODES / 16) * (D / 16) / 4, 128, 0, stream>>>(agg, W, b, Y);
}